// DualPN_21706764714068
// MI455X (gfx1250) — compile-verified
//
#include <hip/hip_runtime.h>
#include <hip/hip_bf16.h>
#include <math.h>

// ---------------- problem constants (from reference setup_inputs) ----------
#define D_IMG  2048
#define D_ATT  312
#define H_ATT  128
#define HH     128
#define TEMP   32.0f
#define THRESH 0.76604444311897803f   // cos(40 deg)
#define NEGINF (-9.0e15f)

typedef __attribute__((ext_vector_type(2))) float v2f;
typedef __attribute__((ext_vector_type(8))) float v8f;

// ===========================================================================
// Generic f32 WMMA GEMM:  C[M,N] = A[M,K] @ B[K,N] (+ bias[N])
// One wave32 computes one 16x16 tile via V_WMMA_F32_16X16X4_F32.
// Lane layout per CDNA5 ISA 7.12.2:
//   A 16x4 : lane = (k>=2)*16 + m ; vgpr0=K even, vgpr1=K odd
//   B 4x16 : lane = (k>=2)*16 + n ; vgpr0=K even, vgpr1=K odd
//   C 16x16: vgpr r -> M = r + 8*(lane>=16), N = lane&15
//
// Edge handling: load indices are CLAMPED (row m only feeds output row m,
// col n only feeds output col n, so garbage lanes are simply never stored).
// Keeps the K-loop branch-free and EXEC all-1s at the WMMA.
// Requires K % 8 == 0 (true for all GEMMs here: 2048, 312, 128, 1000, 624).
// ===========================================================================
__global__ void gemm16_wmma_kernel(const float* __restrict__ A,
                                   const float* __restrict__ B,
                                   const float* __restrict__ bias,
                                   float* __restrict__ C,
                                   int M, int N, int K,
                                   int lda, int ldb, int ldc) {
    const int tilesN = (N + 15) >> 4;
    const int wave   = blockIdx.x * (blockDim.x >> 5) + (threadIdx.x >> 5);
    const int tM = wave / tilesN;
    const int tN = wave - tM * tilesN;
    const int row0 = tM << 4;
    const int col0 = tN << 4;
    if (row0 >= M) return;                 // uniform per-wave exit

    const int lane = threadIdx.x & 31;
    const int half = lane >> 4;            // 0/1 -> K sub-pair within chunk
    const int idx  = lane & 15;
    const int m  = row0 + idx;
    const int n  = col0 + idx;
    const int mc = m < M ? m : (M - 1);    // clamped load indices
    const int nc = n < N ? n : (N - 1);

    const float* pa = A + (size_t)mc * lda + (half << 1);
    const float* pb = B + (size_t)(half << 1) * ldb + nc;
    const size_t ldbs = (size_t)ldb;

    v8f acc0 = {}, acc1 = {};
    for (int k0 = 0; k0 < K; k0 += 8) {
        v2f a0, a1, b0, b1;
        a0.x = pa[0];        a0.y = pa[1];          // contiguous -> b64 load
        a1.x = pa[4];        a1.y = pa[5];
        b0.x = pb[0];        b0.y = pb[ldbs];
        b1.x = pb[4 * ldbs]; b1.y = pb[5 * ldbs];
        pa += 8;
        pb += 8 * ldbs;
        acc0 = __builtin_amdgcn_wmma_f32_16x16x4_f32(
            false, a0, false, b0, (short)0, acc0, false, false);
        acc1 = __builtin_amdgcn_wmma_f32_16x16x4_f32(
            false, a1, false, b1, (short)0, acc1, false, false);
    }

    if (n < N) {
        const float bv = bias ? bias[n] : 0.0f;
        const int mb = row0 + (half << 3);
        #pragma unroll
        for (int r = 0; r < 8; ++r) {
            const int mm = mb + r;
            if (mm < M) C[(size_t)mm * ldc + n] = acc0[r] + acc1[r] + bv;
        }
    }
}

// ===========================================================================
// sim = gn @ gn^T with threshold mask epilogue:
//   sim[i,j] = (s > THRESH) ? s : NEG_INF
// B operand read transposed straight from gn (B[k][n] = gn[n][k]), so both
// operands stream contiguously along K -> all loads are b64.
// ===========================================================================
__global__ void simgemm_wmma_kernel(const float* __restrict__ G,
                                    float* __restrict__ S,
                                    int C, int Kc) {
    const int tiles = (C + 15) >> 4;
    const int wave  = blockIdx.x * (blockDim.x >> 5) + (threadIdx.x >> 5);
    const int tM = wave / tiles;
    const int tN = wave - tM * tiles;
    const int row0 = tM << 4;
    const int col0 = tN << 4;
    if (row0 >= C) return;

    const int lane = threadIdx.x & 31;
    const int half = lane >> 4;
    const int idx  = lane & 15;
    const int m  = row0 + idx;
    const int n  = col0 + idx;
    const int mcl = m < C ? m : (C - 1);
    const int ncl = n < C ? n : (C - 1);

    const float* pa = G + (size_t)mcl * Kc + (half << 1);
    const float* pb = G + (size_t)ncl * Kc + (half << 1);

    v8f acc0 = {}, acc1 = {};
    for (int k0 = 0; k0 < Kc; k0 += 8) {
        v2f a0, a1, b0, b1;
        a0.x = pa[0]; a0.y = pa[1];
        a1.x = pa[4]; a1.y = pa[5];
        b0.x = pb[0]; b0.y = pb[1];
        b1.x = pb[4]; b1.y = pb[5];
        pa += 8;
        pb += 8;
        acc0 = __builtin_amdgcn_wmma_f32_16x16x4_f32(
            false, a0, false, b0, (short)0, acc0, false, false);
        acc1 = __builtin_amdgcn_wmma_f32_16x16x4_f32(
            false, a1, false, b1, (short)0, acc1, false, false);
    }

    if (n < C) {
        const int mb = row0 + (half << 3);
        #pragma unroll
        for (int r = 0; r < 8; ++r) {
            const int mm = mb + r;
            if (mm < C) {
                const float v = acc0[r] + acc1[r];
                S[(size_t)mm * C + n] = (v > THRESH) ? v : NEGINF;
            }
        }
    }
}

// ===========================================================================
// Row L2-normalize g (C rows x 128 cols), in place.  One wave per row.
// ===========================================================================
__global__ void rownorm_kernel(float* __restrict__ G, int C) {
    const int row  = blockIdx.x;
    const int lane = threadIdx.x;          // 32 threads
    if (row >= C) return;
    float* p = G + (size_t)row * H_ATT;
    float v[4];
    float s = 0.0f;
    #pragma unroll
    for (int i = 0; i < 4; ++i) {
        v[i] = p[lane + 32 * i];
        s += v[i] * v[i];
    }
    #pragma unroll
    for (int off = 16; off > 0; off >>= 1) s += __shfl_xor(s, off, 32);
    const float inv = 1.0f / fmaxf(sqrtf(s), 1e-12f);
    #pragma unroll
    for (int i = 0; i < 4; ++i) p[lane + 32 * i] = v[i] * inv;
}

// ===========================================================================
// Row softmax over masked sim, scaled by TEMP, in place. One wave per row.
// ===========================================================================
__global__ void softmax_rows_kernel(float* __restrict__ S, int C) {
    const int row  = blockIdx.x;
    const int lane = threadIdx.x;          // 32 threads
    if (row >= C) return;
    float* p = S + (size_t)row * C;

    float m = -3.0e38f;
    for (int j = lane; j < C; j += 32) m = fmaxf(m, p[j] * TEMP);
    #pragma unroll
    for (int off = 16; off > 0; off >>= 1) m = fmaxf(m, __shfl_xor(m, off, 32));

    float s = 0.0f;
    for (int j = lane; j < C; j += 32) {
        const float e = __expf(p[j] * TEMP - m);
        p[j] = e;
        s += e;
    }
    #pragma unroll
    for (int off = 16; off > 0; off >>= 1) s += __shfl_xor(s, off, 32);

    const float inv = 1.0f / s;
    for (int j = lane; j < C; j += 32) p[j] *= inv;
}

// ===========================================================================
// Relation head: out[b,c] = sum_h relu(img_h[b,h] + proto_h[c,h]) * fc_w[h]
//                           + fc_b            (proto_b already folded in)
// Block = 256 threads = 8 b x 32 c tile; LDS-staged, proto rows padded to
// 129 floats so the stride-128 walk doesn't collide in the 64 LDS banks.
// ===========================================================================
__global__ void relation_kernel(const float* __restrict__ img_h,
                                const float* __restrict__ proto_h,
                                const float* __restrict__ fc_w,
                                const float* __restrict__ fc_b,
                                float* __restrict__ out,
                                int B, int C) {
    __shared__ float s_img[8][HH];
    __shared__ float s_pro[32][HH + 1];
    __shared__ float s_fcw[HH];

    const int t  = threadIdx.x;
    const int b0 = blockIdx.y << 3;
    const int c0 = blockIdx.x << 5;

    if (t < HH) s_fcw[t] = fc_w[t];
    for (int i = t; i < 8 * HH; i += 256) {
        const int bb = i >> 7, hh = i & (HH - 1);
        s_img[bb][hh] = (b0 + bb < B) ? img_h[(size_t)(b0 + bb) * HH + hh] : 0.0f;
    }
    for (int i = t; i < 32 * HH; i += 256) {
        const int cc = i >> 7, hh = i & (HH - 1);
        s_pro[cc][hh] = (c0 + cc < C) ? proto_h[(size_t)(c0 + cc) * HH + hh] : 0.0f;
    }
    __syncthreads();

    const int tc = t & 31, tb = t >> 5;
    float acc = fc_b[0];
    #pragma unroll 8
    for (int h = 0; h < HH; ++h) {
        const float v = s_img[tb][h] + s_pro[tc][h];
        acc = fmaf(fmaxf(v, 0.0f), s_fcw[h], acc);
    }
    const int b = b0 + tb, c = c0 + tc;
    if (b < B && c < C) out[(size_t)b * C + c] = acc;
}

// ===========================================================================
// Host-side orchestration
// ===========================================================================
static inline int gemm_blocks(int M, int N) {
    const int tiles = ((M + 15) / 16) * ((N + 15) / 16);
    return (tiles + 3) / 4;                // 4 waves (128 threads) per block
}

extern "C" void kernel_launch(void* const* d_in, const int* in_sizes, int n_in,
                              void* d_out, int out_size, void* d_ws, size_t ws_size,
                              hipStream_t stream) {
    const float* image_feats = (const float*)d_in[0];   // [B, 2048]
    const float* img_proto   = (const float*)d_in[1];   // [C, 2048]
    const float* attributes  = (const float*)d_in[2];   // [C, 312]
    const float* att_g       = (const float*)d_in[4];   // [312, 128]
    const float* slim_w      = (const float*)d_in[5];   // [2048, 312]
    const float* slim_b      = (const float*)d_in[6];   // [312]
    const float* img_w       = (const float*)d_in[7];   // [2048, 128]
    const float* proto_w     = (const float*)d_in[8];   // [624, 128]
    const float* proto_b     = (const float*)d_in[9];   // [1, 128]
    const float* fc_w        = (const float*)d_in[10];  // [128, 1]
    const float* fc_b        = (const float*)d_in[11];  // [1]
    float* out = (float*)d_out;

    const int B = in_sizes[0] / D_IMG;      // 2048
    const int C = in_sizes[2] / D_ATT;      // 1000

    // workspace carve-up (floats), 256-float aligned
    float* ws = (float*)d_ws;
    size_t off = 0;
    auto carve = [&](size_t n) { float* p = ws + off; off += (n + 255) & ~(size_t)255; return p; };
    float* proto_red = carve((size_t)C * D_ATT);        // [C, 312]
    float* gn        = carve((size_t)C * H_ATT);        // [C, 128]
    float* sim       = carve((size_t)C * C);            // [C, C] -> becomes attn
    float* outs      = carve((size_t)C * 2 * D_ATT);    // [C, 624]
    float* img_h     = carve((size_t)B * HH);           // [B, 128]
    float* proto_h   = carve((size_t)C * HH);           // [C, 128]

    // 1) proto_red = img_proto @ slim_w + slim_b      [C,2048]x[2048,312]
    gemm16_wmma_kernel<<<gemm_blocks(C, D_ATT), 128, 0, stream>>>(
        img_proto, slim_w, slim_b, proto_red, C, D_ATT, D_IMG, D_IMG, D_ATT, D_ATT);

    // 2) g = attributes @ att_g                       [C,312]x[312,128]
    gemm16_wmma_kernel<<<gemm_blocks(C, H_ATT), 128, 0, stream>>>(
        attributes, att_g, nullptr, gn, C, H_ATT, D_ATT, D_ATT, H_ATT, H_ATT);

    // 3) row-normalize g -> gn (in place)
    rownorm_kernel<<<C, 32, 0, stream>>>(gn, C);

    // 4) sim = gn @ gn^T, mask > cos40 else NEG_INF   [C,128]x[128,C]
    simgemm_wmma_kernel<<<gemm_blocks(C, C), 128, 0, stream>>>(gn, sim, C, H_ATT);

    // 5) attn = softmax(sim * T) rowwise (in place)
    softmax_rows_kernel<<<C, 32, 0, stream>>>(sim, C);

    // 6) outs[:, :312]   = attn @ attributes          [C,C]x[C,312]
    gemm16_wmma_kernel<<<gemm_blocks(C, D_ATT), 128, 0, stream>>>(
        sim, attributes, nullptr, outs, C, D_ATT, C, C, D_ATT, 2 * D_ATT);

    // 7) outs[:, 312:]   = attn @ proto_red           [C,C]x[C,312]
    gemm16_wmma_kernel<<<gemm_blocks(C, D_ATT), 128, 0, stream>>>(
        sim, proto_red, nullptr, outs + D_ATT, C, D_ATT, C, C, D_ATT, 2 * D_ATT);

    // 8) img_h = image_feats @ img_w                  [B,2048]x[2048,128]
    gemm16_wmma_kernel<<<gemm_blocks(B, HH), 128, 0, stream>>>(
        image_feats, img_w, nullptr, img_h, B, HH, D_IMG, D_IMG, HH, HH);

    // 9) proto_h = outs @ proto_w + proto_b           [C,624]x[624,128]
    gemm16_wmma_kernel<<<gemm_blocks(C, HH), 128, 0, stream>>>(
        outs, proto_w, proto_b, proto_h, C, HH, 2 * D_ATT, 2 * D_ATT, HH, HH);

    // 10) relation head -> out[B,C]
    dim3 rgrid((C + 31) / 32, (B + 7) / 8);
    relation_kernel<<<rgrid, 256, 0, stream>>>(img_h, proto_h, fc_w, fc_b, out, B, C);
}